// PairwiseRankingLoss_23493471109250
// MI455X (gfx1250) — compile-verified
//
#include <hip/hip_runtime.h>

// PairwiseRankingLoss on MI455X (gfx1250, wave32).
//
//  1) Counting-sort the 8192 items into per-property contiguous positive /
//     negative score segments (pair space shrinks 64M -> sum_p P_p*Q_p ~512K).
//  2) Per property, walk 16x16 pair tiles. d[m][n] = s_pos[m] - s_neg[n] is
//     rank-2, so one exact-fp32 V_WMMA_F32_16X16X4_F32 produces a tile:
//        A(16x4) = [ s_pos | 1 | 0 | 0 ],  B(4x16) = [ 1 ; -s_neg ; 0 ; 0 ]
//     A is hoisted per row-strip; inner loop = clamped B load + cndmask +
//     WMMA + 8 packed hinge ops (co-executing with the matrix pipe).
//     Out-of-range rows/cols padded with +/-1e30 => hinge exactly 0; loads
//     are clamped (not branched) so EXEC stays all-ones (WMMA requirement).
//  3) wave32 shfl_xor reduce -> LDS combine -> one f64 global atomic per
//     workgroup; finalize divides by exact pair count sum_p P_p*Q_p.

typedef __attribute__((ext_vector_type(2))) float v2f;
typedef __attribute__((ext_vector_type(8))) float v8f;

#define NPROPS   32
#define MARGIN_F 1.0f
#define PAD_POS  1e30f
#define PAD_NEG  -1e30f

// ws layout (32-bit words):
//  [0..31]    pos counts        [32..63]  neg counts
//  [64..95]   pos offsets       [96..127] neg offsets
//  [128..191] scatter cursors
//  [192..193] fp64 total accumulator (byte offset 768, 8-aligned)
//  [256 ..]   pos scores (N floats) then neg scores (N floats)

__global__ void prl_init(int* wsi) {
  int i = blockIdx.x * blockDim.x + threadIdx.x;
  if (i < 256) wsi[i] = 0;
}

__global__ void prl_count(const int* __restrict__ labels,
                          const int* __restrict__ pids,
                          int* __restrict__ cnt, int n) {
  int i = blockIdx.x * blockDim.x + threadIdx.x;
  if (i < n) {
    int p = pids[i];
    int bin = labels[i] ? p : (NPROPS + p);
    atomicAdd(&cnt[bin], 1);
  }
}

__global__ void prl_offsets(int* wsi) {
  if (threadIdx.x == 0 && blockIdx.x == 0) {
    int po = 0, no = 0;
    for (int p = 0; p < NPROPS; ++p) {
      wsi[64 + p] = po; po += wsi[p];
      wsi[96 + p] = no; no += wsi[32 + p];
    }
  }
}

__global__ void prl_scatter(const float* __restrict__ scores,
                            const int* __restrict__ labels,
                            const int* __restrict__ pids,
                            int* __restrict__ wsi,
                            float* __restrict__ posS,
                            float* __restrict__ negS, int n) {
  int i = blockIdx.x * blockDim.x + threadIdx.x;
  if (i < n) {
    int p   = pids[i];
    int lab = labels[i];
    int bin = lab ? p : (NPROPS + p);
    int slot = atomicAdd(&wsi[128 + bin], 1);
    if (lab) posS[wsi[64 + p] + slot] = scores[i];
    else     negS[wsi[96 + p] + slot] = scores[i];
  }
}

__global__ void __launch_bounds__(256)
prl_pairs(const int* __restrict__ wsi,
          const float* __restrict__ posS,
          const float* __restrict__ negS,
          double* __restrict__ total) {
  const int prop = blockIdx.x;
  const int P    = wsi[prop];
  const int Q    = wsi[NPROPS + prop];
  const int pOff = wsi[64 + prop];
  const int qOff = wsi[96 + prop];

  const int tilesR = (P + 15) >> 4;
  const int tilesC = (Q + 15) >> 4;

  const int wavesPerBlk = blockDim.x >> 5;             // 8
  const int waveId      = threadIdx.x >> 5;
  const int nWaves      = wavesPerBlk;                 // row strips strided by this

  const int lane = threadIdx.x & 31;
  const int sub  = lane & 15;
  const int half = lane >> 4;       // lanes 16..31 hold the K=2,3 slices -> zero
  const float hz = half ? 0.0f : 1.0f;

  if (tilesR > 0 && tilesC > 0) {
    __builtin_prefetch(posS + pOff, 0, 3);  // global_prefetch_b8 (L2-resident)
    __builtin_prefetch(negS + qOff, 0, 3);
  }

  v8f acc = {};
  // Outer: row strips across the block's 8 waves (no integer division).
  for (int tr = waveId; tr < tilesR; tr += nWaves) {
    const int rI = (tr << 4) + sub;
    // Clamped load + select: always in-bounds (P >= 1 here), no EXEC writes.
    const float spRaw = posS[pOff + min(rI, P - 1)];
    const float sp    = (rI < P) ? spRaw : PAD_POS;

    v2f A;                        // hoisted per strip
    A.x = half ? 0.0f : sp;       // K0 column = s_pos
    A.y = hz;                     // K1 column = 1

    // Inner: column tiles, split across blockIdx.y, uniform trip count.
    for (int tc = blockIdx.y; tc < tilesC; tc += gridDim.y) {
      const int cI = (tc << 4) + sub;
      const float sqRaw = negS[qOff + min(cI, Q - 1)];
      const float sq    = (cI < Q) ? sqRaw : PAD_NEG;

      v2f B;
      B.x = hz;                   // K0 row = 1
      B.y = half ? 0.0f : -sq;    // K1 row = -s_neg
      v8f c = {};
      v8f D = __builtin_amdgcn_wmma_f32_16x16x4_f32(
          /*neg_a=*/false, A, /*neg_b=*/false, B,
          /*c_mod=*/(short)0, c, /*reuse_a=*/false, /*reuse_b=*/false);

#pragma unroll
      for (int r = 0; r < 8; ++r)
        acc[r] += fmaxf(MARGIN_F - D[r], 0.0f);
    }
  }

  float s = acc[0] + acc[1] + acc[2] + acc[3] + acc[4] + acc[5] + acc[6] + acc[7];
#pragma unroll
  for (int off = 16; off > 0; off >>= 1)
    s += __shfl_xor(s, off, 32);            // wave32 butterfly reduce

  __shared__ float wsum[8];
  if (lane == 0) wsum[waveId] = s;
  __syncthreads();
  if (threadIdx.x == 0) {
    float b = 0.0f;
    for (int w = 0; w < wavesPerBlk; ++w) b += wsum[w];
    atomicAdd(total, (double)b);            // global_atomic_add_f64, 1 per workgroup
  }
}

__global__ void prl_final(const int* __restrict__ wsi,
                          const double* __restrict__ total,
                          float* __restrict__ out) {
  if (threadIdx.x == 0 && blockIdx.x == 0) {
    long long np = 0;
    for (int p = 0; p < NPROPS; ++p)
      np += (long long)wsi[p] * (long long)wsi[NPROPS + p];
    out[0] = (np == 0) ? 0.0f : (float)(total[0] / (double)np);
  }
}

extern "C" void kernel_launch(void* const* d_in, const int* in_sizes, int n_in,
                              void* d_out, int out_size, void* d_ws, size_t ws_size,
                              hipStream_t stream) {
  const float* scores = (const float*)d_in[0];
  const int*   labels = (const int*)d_in[1];
  const int*   pids   = (const int*)d_in[2];
  const int    n      = in_sizes[0];

  int*    wsi   = (int*)d_ws;
  float*  wsf   = (float*)d_ws;
  double* total = (double*)((char*)d_ws + 192 * sizeof(int));
  float*  posS  = wsf + 256;
  float*  negS  = wsf + 256 + n;
  float*  out   = (float*)d_out;

  const int blocks = (n + 255) / 256;

  prl_init   <<<1, 256, 0, stream>>>(wsi);
  prl_count  <<<blocks, 256, 0, stream>>>(labels, pids, wsi, n);
  prl_offsets<<<1, 32, 0, stream>>>(wsi);
  prl_scatter<<<blocks, 256, 0, stream>>>(scores, labels, pids, wsi, posS, negS, n);
  dim3 pg(NPROPS, 2);   // 32 properties x 2 column splits, 8 waves each
  prl_pairs  <<<pg, 256, 0, stream>>>(wsi, posS, negS, total);
  prl_final  <<<1, 32, 0, stream>>>(wsi, total, out);
}